// LayerNormMoELayer_46007689675058
// MI455X (gfx1250) — compile-verified
//
#include <hip/hip_runtime.h>
#include <math.h>

// Problem constants (match reference)
#define TOK  8192
#define HID  2048
#define NEXP 8
#define FFD  2048
#define EPSV 1e-5f

// LDS row pitch (floats): 16 K-values + 4 pad -> conflict-free b64 frag loads,
// keeps 16B alignment for float4 cooperative stores (80B row stride).
#define LPITCH 20

typedef __attribute__((ext_vector_type(2))) float v2f;
typedef __attribute__((ext_vector_type(8))) float v8f;

// ---------------------------------------------------------------------------
// Kernel 1: RMSNorm + router logits + softmax + top-2 gates (one block/token)
// ---------------------------------------------------------------------------
__global__ __launch_bounds__(256) void rms_router_kernel(
    const float* __restrict__ x, const float* __restrict__ lnw,
    const float* __restrict__ rw, float* __restrict__ xnorm,
    float* __restrict__ gates)
{
    const int t   = blockIdx.x;
    const int tid = threadIdx.x;
    const float* xr = x + (size_t)t * HID;

    float xv[8];
    float ss = 0.f;
#pragma unroll
    for (int i = 0; i < 8; ++i) {
        const int h = i * 256 + tid;
        xv[i] = xr[h];
        ss += xv[i] * xv[i];
    }
    __shared__ float red[256];
    red[tid] = ss;
    __syncthreads();
    for (int s = 128; s > 0; s >>= 1) {
        if (tid < s) red[tid] += red[tid + s];
        __syncthreads();
    }
    const float scale = rsqrtf(red[0] / (float)HID + EPSV);

    float lg[NEXP];
#pragma unroll
    for (int e = 0; e < NEXP; ++e) lg[e] = 0.f;

#pragma unroll
    for (int i = 0; i < 8; ++i) {
        const int h = i * 256 + tid;
        const float xn = xv[i] * scale * lnw[h];
        xnorm[(size_t)t * HID + h] = xn;
#pragma unroll
        for (int e = 0; e < NEXP; ++e) lg[e] += xn * rw[(size_t)h * NEXP + e];
    }

    __shared__ float slog[NEXP];
    if (tid < NEXP) slog[tid] = 0.f;
    __syncthreads();
#pragma unroll
    for (int e = 0; e < NEXP; ++e) atomicAdd(&slog[e], lg[e]);
    __syncthreads();

    if (tid == 0) {
        float m = slog[0];
        for (int e = 1; e < NEXP; ++e) m = fmaxf(m, slog[e]);
        float p[NEXP];
        float sum = 0.f;
        for (int e = 0; e < NEXP; ++e) { p[e] = __expf(slog[e] - m); sum += p[e]; }
        const float inv = 1.f / sum;
        for (int e = 0; e < NEXP; ++e) p[e] *= inv;
        int i1 = 0;
        for (int e = 1; e < NEXP; ++e) if (p[e] > p[i1]) i1 = e;
        int i2 = (i1 == 0) ? 1 : 0;
        for (int e = 0; e < NEXP; ++e) if (e != i1 && p[e] > p[i2]) i2 = e;
        for (int e = 0; e < NEXP; ++e)
            gates[(size_t)t * NEXP + e] = (e == i1 || e == i2) ? p[e] : 0.f;
    }
}

// ---------------------------------------------------------------------------
// Kernel 2: per-expert fused dual GEMM: act = silu(x@wg) * (x@wu)
// 128x128 block tile; 8 waves as 2(M)x4(N); each wave owns 64x32 =
// 4(M-sub) x 2(N-sub) f32-WMMA D tiles per B matrix.
// ---------------------------------------------------------------------------
__global__ __launch_bounds__(256) void moe_gemm1_kernel(
    const float* __restrict__ xn, const float* __restrict__ wg,
    const float* __restrict__ wu, float* __restrict__ act)
{
    __shared__ float As[128][LPITCH];   // As[m][k]
    __shared__ float Bg[128][LPITCH];   // Bg[n][k]
    __shared__ float Bu[128][LPITCH];

    const int tid  = threadIdx.x;
    const int lane = tid & 31;
    const int wave = tid >> 5;
    const int wm   = wave & 1;   // 2 tiles of 64 rows
    const int wn   = wave >> 1;  // 4 tiles of 32 cols
    const int lh   = lane >> 4;  // lane half (K select)
    const int lm   = lane & 15;

    const int t0 = blockIdx.y * 128;
    const int n0 = blockIdx.x * 128;

    v8f accg[4][2], accu[4][2];
    const v8f vzero = {};
#pragma unroll
    for (int ms = 0; ms < 4; ++ms)
#pragma unroll
        for (int ns = 0; ns < 2; ++ns) { accg[ms][ns] = vzero; accu[ms][ns] = vzero; }

    for (int k0 = 0; k0 < HID; k0 += 16) {
        // Cooperative loads: A = 128x16 (contig along K), B = 16x128 (contig along N)
#pragma unroll
        for (int r = 0; r < 2; ++r) {
            const int f = tid + r * 256;
            const int m = f >> 2, q = f & 3;
            *(float4*)(&As[m][q * 4]) =
                *(const float4*)(&xn[(size_t)(t0 + m) * HID + k0 + q * 4]);
            const int k = f >> 5, c = f & 31;
            const float4 gv = *(const float4*)(&wg[(size_t)(k0 + k) * FFD + n0 + c * 4]);
            Bg[c * 4 + 0][k] = gv.x; Bg[c * 4 + 1][k] = gv.y;
            Bg[c * 4 + 2][k] = gv.z; Bg[c * 4 + 3][k] = gv.w;
            const float4 uv = *(const float4*)(&wu[(size_t)(k0 + k) * FFD + n0 + c * 4]);
            Bu[c * 4 + 0][k] = uv.x; Bu[c * 4 + 1][k] = uv.y;
            Bu[c * 4 + 2][k] = uv.z; Bu[c * 4 + 3][k] = uv.w;
        }
        __syncthreads();

#pragma unroll
        for (int kk = 0; kk < 4; ++kk) {
            const int kb = kk * 4 + 2 * lh;  // ISA A/B layout: K = v + 2*(lane/16)
            v2f a[4], bgf[2], buf[2];
#pragma unroll
            for (int ms = 0; ms < 4; ++ms)
                a[ms] = *(const v2f*)(&As[wm * 64 + ms * 16 + lm][kb]);
#pragma unroll
            for (int ns = 0; ns < 2; ++ns) {
                bgf[ns] = *(const v2f*)(&Bg[wn * 32 + ns * 16 + lm][kb]);
                buf[ns] = *(const v2f*)(&Bu[wn * 32 + ns * 16 + lm][kb]);
            }
#pragma unroll
            for (int ms = 0; ms < 4; ++ms)
#pragma unroll
                for (int ns = 0; ns < 2; ++ns) {
                    accg[ms][ns] = __builtin_amdgcn_wmma_f32_16x16x4_f32(
                        false, a[ms], false, bgf[ns], (short)0, accg[ms][ns], false, false);
                    accu[ms][ns] = __builtin_amdgcn_wmma_f32_16x16x4_f32(
                        false, a[ms], false, buf[ns], (short)0, accu[ms][ns], false, false);
                }
        }
        __syncthreads();
    }

    // SwiGLU epilogue: D-tile element (v, lane) -> row = v + 8*(lane/16)
#pragma unroll
    for (int ms = 0; ms < 4; ++ms)
#pragma unroll
        for (int ns = 0; ns < 2; ++ns) {
            const int col = n0 + wn * 32 + ns * 16 + lm;
#pragma unroll
            for (int v = 0; v < 8; ++v) {
                const int row = t0 + wm * 64 + ms * 16 + v + 8 * lh;
                const float g = accg[ms][ns][v];
                const float u = accu[ms][ns][v];
                act[(size_t)row * FFD + col] = (g / (1.f + __expf(-g))) * u;
            }
        }
}

// ---------------------------------------------------------------------------
// Kernel 3: per-expert down-proj + gate-weighted accumulate into out
// ---------------------------------------------------------------------------
__global__ __launch_bounds__(256) void moe_gemm2_kernel(
    const float* __restrict__ act, const float* __restrict__ wd,
    const float* __restrict__ gates, float* __restrict__ out,
    int expert, int accumulate)
{
    __shared__ float As[128][LPITCH];   // As[m][k]
    __shared__ float Bs[128][LPITCH];   // Bs[n][k]
    __shared__ float gt[128];

    const int tid  = threadIdx.x;
    const int lane = tid & 31;
    const int wave = tid >> 5;
    const int wm   = wave & 1;
    const int wn   = wave >> 1;
    const int lh   = lane >> 4;
    const int lm   = lane & 15;

    const int t0 = blockIdx.y * 128;
    const int n0 = blockIdx.x * 128;

    if (tid < 128) gt[tid] = gates[(size_t)(t0 + tid) * NEXP + expert];

    v8f acc[4][2];
    const v8f vzero = {};
#pragma unroll
    for (int ms = 0; ms < 4; ++ms)
#pragma unroll
        for (int ns = 0; ns < 2; ++ns) acc[ms][ns] = vzero;

    for (int k0 = 0; k0 < FFD; k0 += 16) {
#pragma unroll
        for (int r = 0; r < 2; ++r) {
            const int f = tid + r * 256;
            const int m = f >> 2, q = f & 3;
            *(float4*)(&As[m][q * 4]) =
                *(const float4*)(&act[(size_t)(t0 + m) * FFD + k0 + q * 4]);
            const int k = f >> 5, c = f & 31;
            const float4 dv = *(const float4*)(&wd[(size_t)(k0 + k) * HID + n0 + c * 4]);
            Bs[c * 4 + 0][k] = dv.x; Bs[c * 4 + 1][k] = dv.y;
            Bs[c * 4 + 2][k] = dv.z; Bs[c * 4 + 3][k] = dv.w;
        }
        __syncthreads();

#pragma unroll
        for (int kk = 0; kk < 4; ++kk) {
            const int kb = kk * 4 + 2 * lh;
            v2f a[4], b[2];
#pragma unroll
            for (int ms = 0; ms < 4; ++ms)
                a[ms] = *(const v2f*)(&As[wm * 64 + ms * 16 + lm][kb]);
#pragma unroll
            for (int ns = 0; ns < 2; ++ns)
                b[ns] = *(const v2f*)(&Bs[wn * 32 + ns * 16 + lm][kb]);
#pragma unroll
            for (int ms = 0; ms < 4; ++ms)
#pragma unroll
                for (int ns = 0; ns < 2; ++ns)
                    acc[ms][ns] = __builtin_amdgcn_wmma_f32_16x16x4_f32(
                        false, a[ms], false, b[ns], (short)0, acc[ms][ns], false, false);
        }
        __syncthreads();
    }

#pragma unroll
    for (int ms = 0; ms < 4; ++ms)
#pragma unroll
        for (int ns = 0; ns < 2; ++ns) {
            const int col = n0 + wn * 32 + ns * 16 + lm;
#pragma unroll
            for (int v = 0; v < 8; ++v) {
                const int row = wm * 64 + ms * 16 + v + 8 * lh;
                const size_t idx = (size_t)(t0 + row) * HID + col;
                const float val = gt[row] * acc[ms][ns][v];
                if (accumulate) out[idx] += val;
                else            out[idx]  = val;
            }
        }
}

// ---------------------------------------------------------------------------
// Host launcher
// ---------------------------------------------------------------------------
extern "C" void kernel_launch(void* const* d_in, const int* in_sizes, int n_in,
                              void* d_out, int out_size, void* d_ws, size_t ws_size,
                              hipStream_t stream)
{
    (void)in_sizes; (void)n_in; (void)out_size; (void)ws_size;
    const float* x   = (const float*)d_in[0];
    const float* lnw = (const float*)d_in[1];
    const float* rw  = (const float*)d_in[2];
    const float* wg  = (const float*)d_in[3];
    const float* wu  = (const float*)d_in[4];
    const float* wd  = (const float*)d_in[5];
    float* out = (float*)d_out;

    char* ws = (char*)d_ws;
    float* xnorm = (float*)ws;                                          // 64 MB
    float* act   = (float*)(ws + (size_t)TOK * HID * sizeof(float));    // 64 MB
    float* gates = (float*)(ws + (size_t)TOK * HID * sizeof(float)
                               + (size_t)TOK * FFD * sizeof(float));    // 256 KB

    rms_router_kernel<<<TOK, 256, 0, stream>>>(x, lnw, rw, xnorm, gates);

    for (int e = 0; e < NEXP; ++e) {
        moe_gemm1_kernel<<<dim3(FFD / 128, TOK / 128), 256, 0, stream>>>(
            xnorm, wg + (size_t)e * HID * FFD, wu + (size_t)e * HID * FFD, act);
        moe_gemm2_kernel<<<dim3(HID / 128, TOK / 128), 256, 0, stream>>>(
            act, wd + (size_t)e * FFD * HID, gates, out, e, e > 0 ? 1 : 0);
    }
}